// Autoencoder_48206712930890
// MI455X (gfx1250) — compile-verified
//
#include <hip/hip_runtime.h>
#include <stdint.h>

// ---------------------------------------------------------------------------
// Volume ray-marcher for MI455X (gfx1250, wave32).
//  * 64 MB template fits in 192 MB L2 -> L2-resident gather kernel.
//  * Pre-pass transposes template (B,4,D,D,D) -> (B,D,D,D)x float4 in d_ws,
//    turning 32 scalar gathers/step into 8x global_load_b128.
//  * Ray-direction einsum done with V_WMMA_F32_16X16X4_F32 (the one real
//    matmul in the workload): A' = R^T padded to 16x4, B' = 16 pixel dirs,
//    D' rows 0..2 = rd components with pixel == lane.
//  * threefry2x32-20 matches jax.random; 73 step keys cached in LDS.
//  * global_prefetch_b8 of next step's base corner.
// ---------------------------------------------------------------------------

#define Bn 2
#define Hh 384
#define Ww 384
#define Dvox 128
#define DT_C 0.06f
#define JIT_C 0.01f
#define NSTEPS 72
#define NPIX (Bn * Hh * Ww)
#define VOX ((size_t)Dvox * Dvox * Dvox)

typedef __attribute__((ext_vector_type(2))) float v2f;
typedef __attribute__((ext_vector_type(8))) float v8f;

// ----------------------------- threefry2x32 --------------------------------
__device__ __forceinline__ uint2 tf2x32(uint2 key, uint2 ctr) {
  uint32_t ks0 = key.x, ks1 = key.y, ks2 = key.x ^ key.y ^ 0x1BD11BDAu;
  uint32_t x0 = ctr.x + ks0, x1 = ctr.y + ks1;
#define TF_R(r) { x0 += x1; x1 = (x1 << (r)) | (x1 >> (32 - (r))); x1 ^= x0; }
  TF_R(13) TF_R(15) TF_R(26) TF_R(6)  x0 += ks1; x1 += ks2 + 1u;
  TF_R(17) TF_R(29) TF_R(16) TF_R(24) x0 += ks2; x1 += ks0 + 2u;
  TF_R(13) TF_R(15) TF_R(26) TF_R(6)  x0 += ks0; x1 += ks1 + 3u;
  TF_R(17) TF_R(29) TF_R(16) TF_R(24) x0 += ks1; x1 += ks2 + 4u;
  TF_R(13) TF_R(15) TF_R(26) TF_R(6)  x0 += ks2; x1 += ks0 + 5u;
#undef TF_R
  return make_uint2(x0, x1);
}

// jax random_bits layout: counters split in two halves, out = [x0 | x1]
__device__ __forceinline__ uint32_t rnd_bits(uint2 key, uint32_t j) {
  const uint32_t half = NPIX / 2;
  uint32_t c0 = (j < half) ? j : (j - half);
  uint2 o = tf2x32(key, make_uint2(c0, c0 + half));
  return (j < half) ? o.x : o.y;
}

__device__ __forceinline__ float u01f(uint32_t b) {
  return __uint_as_float((b >> 9) | 0x3f800000u) - 1.0f;  // [0,1)
}

// Giles (2012) single-precision erfinv
__device__ __forceinline__ float erfinv_f(float x) {
  float w = -logf((1.0f - x) * (1.0f + x));
  float p;
  if (w < 5.0f) {
    w = w - 2.5f;
    p = 2.81022636e-08f;
    p = 3.43273939e-07f + p * w;
    p = -3.5233877e-06f + p * w;
    p = -4.39150654e-06f + p * w;
    p = 0.00021858087f + p * w;
    p = -0.00125372503f + p * w;
    p = -0.00417768164f + p * w;
    p = 0.246640727f + p * w;
    p = 1.50140941f + p * w;
  } else {
    w = sqrtf(w) - 3.0f;
    p = -0.000200214257f;
    p = 0.000100950558f + p * w;
    p = 0.00134934322f + p * w;
    p = -0.00367342844f + p * w;
    p = 0.00573950773f + p * w;
    p = -0.0076224613f + p * w;
    p = 0.00943887047f + p * w;
    p = 1.00167406f + p * w;
    p = 2.83297682f + p * w;
  }
  return p * x;
}

// keys[i] = fold_in(key(42), i) = threefry((0,42),(0,i)); uniform over grid.
__global__ void make_keys(uint2* __restrict__ keys) {
  int i = threadIdx.x;
  if (i <= NSTEPS) keys[i] = tf2x32(make_uint2(0u, 42u), make_uint2(0u, (uint32_t)i));
}

// (B,4,D,D,D) channel-major -> (B,D^3) float4 voxel-major.  All four channel
// streams and the float4 store are fully coalesced.
__global__ __launch_bounds__(256) void vol_transpose(const float* __restrict__ t,
                                                     float4* __restrict__ v) {
  size_t i = (size_t)blockIdx.x * 256 + threadIdx.x;  // 0 .. B*VOX-1
  size_t b = i / VOX, r = i - b * VOX;
  const float* p = t + b * 4 * VOX + r;
  v[i] = make_float4(p[0], p[VOX], p[2 * VOX], p[3 * VOX]);
}

__device__ __forceinline__ float4 lerp4(float4 a, float4 b, float w) {
  float iw = 1.0f - w;
  return make_float4(a.x * iw + b.x * w, a.y * iw + b.y * w,
                     a.z * iw + b.z * w, a.w * iw + b.w * w);
}

template <bool V4>
__device__ __forceinline__ float4 fetch_vox(const float4* __restrict__ vol,
                                            const float* __restrict__ tmpl,
                                            size_t bvox, size_t off) {
  if (V4) return vol[bvox + off];
  const float* p = tmpl + 4 * bvox + off;
  return make_float4(p[0], p[VOX], p[2 * VOX], p[3 * VOX]);
}

template <bool V4>
__global__ __launch_bounds__(256) void raymarch(
    const float* __restrict__ camR, const float* __restrict__ camP,
    const float* __restrict__ foc, const float* __restrict__ pp,
    const float* __restrict__ pix, const float* __restrict__ tmpl,
    const float4* __restrict__ vol, const float* __restrict__ bg,
    const uint2* __restrict__ gkeys, float* __restrict__ out) {
  __shared__ uint2 keys[NSTEPS + 1];
  if (threadIdx.x <= NSTEPS) keys[threadIdx.x] = gkeys[threadIdx.x];
  __syncthreads();

  const int idx = blockIdx.x * 256 + threadIdx.x;  // flat (b,h,w) pixel id
  const int lane = threadIdx.x & 31;
  const int b = idx / (Hh * Ww);
  const int hw = idx - b * (Hh * Ww);
  const int h = hw / Ww, w = hw - h * Ww;

  const float fx = foc[2 * b], fy = foc[2 * b + 1];
  const float ppx = pp[2 * b], ppy = pp[2 * b + 1];
  const float cpx = camP[3 * b], cpy = camP[3 * b + 1], cpz = camP[3 * b + 2];
  const float* Rm = camR + 9 * b;

  // ---- rd = R^T * dirs via V_WMMA_F32_16X16X4_F32 (2 tiles of 16 pixels) --
  // D'[n][m] = sum_k R[k][n] * dirs[m][k]  ->  pixel m == lane, comp n == row.
  const int wbase = idx & ~31;
  const int n = lane & 15;
  v2f A;
  if (lane < 16) {                     // A 16x4: v[0]=K0, v[1]=K1 (lanes 0-15)
    A.x = (n < 3) ? Rm[0 * 3 + n] : 0.0f;  // A'[n][0] = R[0][n]
    A.y = (n < 3) ? Rm[1 * 3 + n] : 0.0f;  // A'[n][1] = R[1][n]
  } else {                             // lanes 16-31: v[0]=K2, v[1]=K3
    A.x = (n < 3) ? Rm[2 * 3 + n] : 0.0f;  // A'[n][2] = R[2][n]
    A.y = 0.0f;                            // A'[n][3] = 0
  }
  float d0x, d0y, d1x, d1y;
  { int p0 = wbase + n;      d0x = (pix[2 * p0] - ppx) / fx; d0y = (pix[2 * p0 + 1] - ppy) / fy; }
  { int p1 = wbase + 16 + n; d1x = (pix[2 * p1] - ppx) / fx; d1y = (pix[2 * p1 + 1] - ppy) / fy; }
  v2f Bm0, Bm1;  // B 4x16: rows striped over lanes; K2 row = homogeneous 1
  Bm0.x = (lane < 16) ? d0x : 1.0f;
  Bm0.y = (lane < 16) ? d0y : 0.0f;
  Bm1.x = (lane < 16) ? d1x : 1.0f;
  Bm1.y = (lane < 16) ? d1y : 0.0f;
  v8f Cz = {};
  v8f Dm0 = __builtin_amdgcn_wmma_f32_16x16x4_f32(false, A, false, Bm0, (short)0, Cz, false, false);
  v8f Dm1 = __builtin_amdgcn_wmma_f32_16x16x4_f32(false, A, false, Bm1, (short)0, Cz, false, false);
  float sx = __shfl(Dm1[0], n, 32);  // upper half-wave pulls tile-1 results
  float sy = __shfl(Dm1[1], n, 32);
  float sz = __shfl(Dm1[2], n, 32);
  float rdx = (lane < 16) ? Dm0[0] : sx;
  float rdy = (lane < 16) ? Dm0[1] : sy;
  float rdz = (lane < 16) ? Dm0[2] : sz;
  float rin = 1.0f / sqrtf(rdx * rdx + rdy * rdy + rdz * rdz);
  rdx *= rin; rdy *= rin; rdz *= rin;

  // ---- unit-cube intersection --------------------------------------------
  float t1x = (-1.0f - cpx) / rdx, t2x = (1.0f - cpx) / rdx;
  float t1y = (-1.0f - cpy) / rdy, t2y = (1.0f - cpy) / rdy;
  float t1z = (-1.0f - cpz) / rdz, t2z = (1.0f - cpz) / rdz;
  float tmin = fmaxf(fminf(t1x, t2x), fmaxf(fminf(t1y, t2y), fminf(t1z, t2z)));
  float tmax = fminf(fmaxf(t1x, t2x), fminf(fmaxf(t1y, t2y), fmaxf(t1z, t2z)));
  float t = (tmin < tmax) ? fmaxf(tmin, 0.0f) : 0.0f;
  t -= DT_C * u01f(rnd_bits(keys[0], (uint32_t)idx));

  float px_ = cpx + rdx * t, py_ = cpy + rdy * t, pz_ = cpz + rdz * t;
  float accR = 0.0f, accG = 0.0f, accB = 0.0f, accA = 0.0f;
  const size_t bvox = (size_t)b * VOX;
  const float sg = (Dvox - 1) * 0.5f;
  const float lo = -0.99999994f;  // nextafter(-1, 0)

#pragma unroll 2
  for (int i = 0; i < NSTEPS; ++i) {
    float valid = (px_ > -1.0f && px_ < 1.0f && py_ > -1.0f && py_ < 1.0f &&
                   pz_ > -1.0f && pz_ < 1.0f) ? 1.0f : 0.0f;
    float ix = fminf(fmaxf((px_ + 1.0f) * sg, 0.0f), (float)(Dvox - 1));
    float iy = fminf(fmaxf((py_ + 1.0f) * sg, 0.0f), (float)(Dvox - 1));
    float iz = fminf(fmaxf((pz_ + 1.0f) * sg, 0.0f), (float)(Dvox - 1));
    int x0 = (int)ix, y0 = (int)iy, z0 = (int)iz;
    int x1 = (x0 < Dvox - 1) ? x0 + 1 : Dvox - 1;
    int y1 = (y0 < Dvox - 1) ? y0 + 1 : Dvox - 1;
    int z1 = (z0 < Dvox - 1) ? z0 + 1 : Dvox - 1;
    float wx = ix - (float)x0, wy = iy - (float)y0, wz = iz - (float)z0;
    size_t r00 = ((size_t)z0 * Dvox + y0) * Dvox;
    size_t r01 = ((size_t)z0 * Dvox + y1) * Dvox;
    size_t r10 = ((size_t)z1 * Dvox + y0) * Dvox;
    size_t r11 = ((size_t)z1 * Dvox + y1) * Dvox;
    float4 v000 = fetch_vox<V4>(vol, tmpl, bvox, r00 + x0);
    float4 v001 = fetch_vox<V4>(vol, tmpl, bvox, r00 + x1);
    float4 v010 = fetch_vox<V4>(vol, tmpl, bvox, r01 + x0);
    float4 v011 = fetch_vox<V4>(vol, tmpl, bvox, r01 + x1);
    float4 v100 = fetch_vox<V4>(vol, tmpl, bvox, r10 + x0);
    float4 v101 = fetch_vox<V4>(vol, tmpl, bvox, r10 + x1);
    float4 v110 = fetch_vox<V4>(vol, tmpl, bvox, r11 + x0);
    float4 v111 = fetch_vox<V4>(vol, tmpl, bvox, r11 + x1);
    float4 c00 = lerp4(v000, v001, wx);
    float4 c01 = lerp4(v010, v011, wx);
    float4 c10 = lerp4(v100, v101, wx);
    float4 c11 = lerp4(v110, v111, wx);
    float4 smp = lerp4(lerp4(c00, c01, wy), lerp4(c10, c11, wy), wz);

    // step = DT * exp(0.01 * N(0,1)); N via sqrt(2)*erfinv(U(lo,1))
    float f = u01f(rnd_bits(keys[i + 1], (uint32_t)idx));
    float u = lo + f * (1.0f - lo);
    float noise = 1.41421356f * erfinv_f(u);
    float step = DT_C * expf(JIT_C * noise);

    float contrib = (fminf(accA + smp.w * step, 1.0f) - accA) * valid;
    accR += smp.x * contrib;
    accG += smp.y * contrib;
    accB += smp.z * contrib;
    accA += contrib;
    px_ += rdx * step; py_ += rdy * step; pz_ += rdz * step;

    if (V4 && (i + 1 < NSTEPS)) {  // prefetch next step's base corner
      float jx = fminf(fmaxf((px_ + 1.0f) * sg, 0.0f), (float)(Dvox - 1));
      float jy = fminf(fmaxf((py_ + 1.0f) * sg, 0.0f), (float)(Dvox - 1));
      float jz = fminf(fmaxf((pz_ + 1.0f) * sg, 0.0f), (float)(Dvox - 1));
      size_t noff = (((size_t)(int)jz * Dvox + (int)jy) * Dvox + (int)jx);
      __builtin_prefetch((const void*)(vol + bvox + noff), 0, 1);
    }
  }

  const size_t plane = (size_t)Hh * Ww;
  const size_t bbase = (size_t)b * 3 * plane + (size_t)h * Ww + w;
  out[bbase + 0 * plane] = accR + (1.0f - accA) * fmaxf(bg[bbase + 0 * plane], 0.0f);
  out[bbase + 1 * plane] = accG + (1.0f - accA) * fmaxf(bg[bbase + 1 * plane], 0.0f);
  out[bbase + 2 * plane] = accB + (1.0f - accA) * fmaxf(bg[bbase + 2 * plane], 0.0f);
}

extern "C" void kernel_launch(void* const* d_in, const int* in_sizes, int n_in,
                              void* d_out, int out_size, void* d_ws, size_t ws_size,
                              hipStream_t stream) {
  (void)in_sizes; (void)n_in; (void)out_size;
  const float* camR = (const float*)d_in[0];
  const float* camP = (const float*)d_in[1];
  const float* foc  = (const float*)d_in[2];
  const float* pp   = (const float*)d_in[3];
  const float* pix  = (const float*)d_in[4];
  const float* tmpl = (const float*)d_in[5];
  const float* bg   = (const float*)d_in[6];
  float* out = (float*)d_out;

  const size_t volBytes = (size_t)Bn * VOX * sizeof(float4);  // 64 MiB
  const size_t keyBytes = (NSTEPS + 1) * sizeof(uint2);
  const bool useV4 = ws_size >= volBytes + keyBytes;
  float4* vol = useV4 ? (float4*)d_ws : nullptr;
  uint2* keys = useV4 ? (uint2*)((char*)d_ws + volBytes) : (uint2*)d_ws;

  make_keys<<<1, 96, 0, stream>>>(keys);
  if (useV4) {
    vol_transpose<<<(int)((Bn * VOX) / 256), 256, 0, stream>>>(tmpl, vol);
    raymarch<true><<<NPIX / 256, 256, 0, stream>>>(camR, camP, foc, pp, pix,
                                                   tmpl, vol, bg, keys, out);
  } else {
    raymarch<false><<<NPIX / 256, 256, 0, stream>>>(camR, camP, foc, pp, pix,
                                                    tmpl, vol, bg, keys, out);
  }
}